// CircleLoss_38628935860375
// MI455X (gfx1250) — compile-verified
//
#include <hip/hip_runtime.h>
#include <hip/hip_bf16.h>
#include <math.h>

typedef __attribute__((ext_vector_type(2))) float v2f;
typedef __attribute__((ext_vector_type(8))) float v8f;

#define NB 8192
#define ND 128
#define SCALE_S 32.0f
#define MARGIN_M 0.25f

// ---------------------------------------------------------------------------
// Kernel 1: L2-normalize each row of feats (8192 x 128). One wave per row.
// ---------------------------------------------------------------------------
__global__ void __launch_bounds__(256) circle_normalize(const float* __restrict__ feats,
                                                        float* __restrict__ fn) {
    const int wave = threadIdx.x >> 5;
    const int lane = threadIdx.x & 31;
    const int row  = blockIdx.x * 8 + wave;

    const float4* src = reinterpret_cast<const float4*>(feats + (size_t)row * ND);
    float4 v = src[lane];                         // 4 contiguous floats per lane
    float ss = v.x * v.x + v.y * v.y + v.z * v.z + v.w * v.w;
#pragma unroll
    for (int m = 16; m >= 1; m >>= 1) ss += __shfl_xor(ss, m, 32);
    float inv = 1.0f / sqrtf(ss);
    float4 o; o.x = v.x * inv; o.y = v.y * inv; o.z = v.z * inv; o.w = v.w * inv;
    reinterpret_cast<float4*>(fn + (size_t)row * ND)[lane] = o;
}

// ---------------------------------------------------------------------------
// Kernel 2: fused sim-tile GEMM (V_WMMA_F32_16X16X4_F32) + circle-loss
// row reductions. One workgroup owns a 16-row strip; its 8 waves stride
// across the 512 column tiles, TWO tiles per iteration so each wave keeps
// two independent WMMA accumulator chains in flight (hides the serial
// D->C accumulate latency of the K=4 f32 WMMA). sim is never materialized.
//
// A 16x4 F32 fragment layout (ISA 7.12.2): lane l -> M = l%16,
//   K = 4*kb + 2*(l/16) + v  (v = component of the v2f operand).
// B 4x16 mirrors with N = l%16; since sim = f f^T, A and B tiles load with
// the identical lane->(row,k) pattern from the same normalized matrix.
// C/D layout: component r -> M = r + 8*(l/16), N = l%16.
// ---------------------------------------------------------------------------
__global__ void __launch_bounds__(256) circle_strip(const float* __restrict__ fn,
                                                    const int*   __restrict__ labels,
                                                    float* __restrict__ lossP,
                                                    float* __restrict__ lossN) {
    __shared__ float sP[16];
    __shared__ float sN[16];

    const int tid  = threadIdx.x;
    const int wave = tid >> 5;
    const int lane = tid & 31;
    const int l16  = lane & 15;
    const int hi   = lane >> 4;           // which 16-lane half

    if (tid < 16) { sP[tid] = 0.0f; sN[tid] = 0.0f; }
    __syncthreads();

    const int rowBase = blockIdx.x * 16;

    // Cache this strip's A fragments in registers: 32 x v2f = 64 VGPRs.
    const float* fa = fn + (size_t)(rowBase + l16) * ND + hi * 2;
    v2f a[32];
#pragma unroll
    for (int kb = 0; kb < 32; ++kb)
        a[kb] = *reinterpret_cast<const v2f*>(fa + kb * 4);

    // Row labels for this lane's 8 output rows (C layout: M = r + 8*hi).
    int labI[8];
#pragma unroll
    for (int r = 0; r < 8; ++r) labI[r] = labels[rowBase + r + 8 * hi];

    float pP[8], pN[8];
#pragma unroll
    for (int r = 0; r < 8; ++r) { pP[r] = 0.0f; pN[r] = 0.0f; }

    // 512 column tiles; each wave takes a pair of adjacent tiles per pass.
    for (int ct = wave * 2; ct < NB / 16; ct += 16) {
        const int j0    = ct * 16 + l16;          // lane's column in tile 0
        const int j1    = j0 + 16;                // lane's column in tile 1
        const int labJ0 = labels[j0];
        const int labJ1 = labels[j1];

        const float* fb0 = fn + (size_t)j0 * ND + hi * 2;
        const float* fb1 = fn + (size_t)j1 * ND + hi * 2;

        v8f c0 = {};
        v8f c1 = {};
#pragma unroll
        for (int kb = 0; kb < 32; ++kb) {
            v2f b0 = *reinterpret_cast<const v2f*>(fb0 + kb * 4);
            v2f b1 = *reinterpret_cast<const v2f*>(fb1 + kb * 4);
            // 8 args: (neg_a, A, neg_b, B, c_mod, C, reuse_a, reuse_b)
            c0 = __builtin_amdgcn_wmma_f32_16x16x4_f32(
                     false, a[kb], false, b0, (short)0, c0, false, false);
            c1 = __builtin_amdgcn_wmma_f32_16x16x4_f32(
                     false, a[kb], false, b1, (short)0, c1, false, false);
        }

#pragma unroll
        for (int t = 0; t < 2; ++t) {
            const v8f&  c    = t ? c1 : c0;
            const int   j    = t ? j1 : j0;
            const int   labJ = t ? labJ1 : labJ0;
#pragma unroll
            for (int r = 0; r < 8; ++r) {
                const float s    = c[r];
                const int   i    = rowBase + r + 8 * hi;
                const bool  same = (labI[r] == labJ);

                // positive pairs: exp(-S * relu(1+m-s) * (s-(1-m)))
                float ap = fmaxf(1.0f + MARGIN_M - s, 0.0f);
                float ep = __expf(-SCALE_S * ap * (s - (1.0f - MARGIN_M)));
                if (same && (i != j)) pP[r] += ep;

                // negative pairs: exp(S * relu(s+m) * (s-m))
                float an = fmaxf(s + MARGIN_M, 0.0f);
                float en = __expf(SCALE_S * an * (s - MARGIN_M));
                if (!same) pN[r] += en;
            }
        }
    }

    // Reduce over the 16 N-lanes of each half; lanes 0 / 16 own the row sums.
#pragma unroll
    for (int r = 0; r < 8; ++r) {
        float vp = pP[r], vn = pN[r];
#pragma unroll
        for (int m = 1; m < 16; m <<= 1) {
            vp += __shfl_xor(vp, m, 32);
            vn += __shfl_xor(vn, m, 32);
        }
        if (l16 == 0) {
            atomicAdd(&sP[r + 8 * hi], vp);   // ds_add_f32 across 8 waves
            atomicAdd(&sN[r + 8 * hi], vn);
        }
    }
    __syncthreads();

    if (tid < 16) {
        lossP[rowBase + tid] = sP[tid];
        lossN[rowBase + tid] = sN[tid];
    }
}

// ---------------------------------------------------------------------------
// Kernel 3: loss = mean(log1p(lossP * lossN)) over 8192 rows.
// ---------------------------------------------------------------------------
__global__ void __launch_bounds__(256) circle_finish(const float* __restrict__ lossP,
                                                     const float* __restrict__ lossN,
                                                     float* __restrict__ out) {
    __shared__ float red[256];
    const int tid = threadIdx.x;
    float acc = 0.0f;
    for (int i = tid; i < NB; i += 256)
        acc += log1pf(lossP[i] * lossN[i]);
    red[tid] = acc;
    __syncthreads();
    for (int s2 = 128; s2 > 0; s2 >>= 1) {
        if (tid < s2) red[tid] += red[tid + s2];
        __syncthreads();
    }
    if (tid == 0) out[0] = red[0] / (float)NB;
}

// ---------------------------------------------------------------------------
extern "C" void kernel_launch(void* const* d_in, const int* in_sizes, int n_in,
                              void* d_out, int out_size, void* d_ws, size_t ws_size,
                              hipStream_t stream) {
    const float* feats  = (const float*)d_in[0];
    const int*   labels = (const int*)d_in[1];

    float* fn    = (float*)d_ws;                  // 8192*128 floats = 4 MB
    float* lossP = fn + (size_t)NB * ND;          // 8192 floats
    float* lossN = lossP + NB;                    // 8192 floats

    circle_normalize<<<NB / 8, 256, 0, stream>>>(feats, fn);
    circle_strip<<<NB / 16, 256, 0, stream>>>(fn, labels, lossP, lossN);
    circle_finish<<<1, 256, 0, stream>>>(lossP, lossN, (float*)d_out);
}